// NCDE_Decoder_52037823758377
// MI455X (gfx1250) — compile-verified
//
#include <hip/hip_runtime.h>

// Problem constants (from reference): B=4096, L=256, C=16, H=128, O=4
#define BATCH   4096
#define LSEQ    256
#define TSTEPS  255
#define CDIM    16
#define HDIM    128
#define ODIM    4
#define TILE_B  16
#define THREADS 256
#define WAVES   8
#define TPW     16      // hidden-unit tiles per wave (128 / 8)
#define ROWP    132     // padded LDS row (floats): bank skew of 4, rows stay 16B-aligned

typedef float v2f __attribute__((ext_vector_type(2)));
typedef float v4f __attribute__((ext_vector_type(4)));
typedef float v8f __attribute__((ext_vector_type(8)));

__device__ __forceinline__ float fast_tanh(float x) {
#if __has_builtin(__builtin_amdgcn_tanhf)
  return __builtin_amdgcn_tanhf(x);           // gfx1250 v_tanh_f32
#else
  float t = __builtin_amdgcn_exp2f(x * 2.88539008177792681f); // 2*log2(e)
  float r = __builtin_amdgcn_rcpf(t + 1.0f);
  return __builtin_fmaf(-2.0f, r, 1.0f);
#endif
}

__launch_bounds__(THREADS)
__global__ void ncde_decoder_kernel(const float* __restrict__ coeffs,
                                    const float* __restrict__ W_init,
                                    const float* __restrict__ b_init,
                                    const float* __restrict__ W1,
                                    const float* __restrict__ b1,
                                    const float* __restrict__ W2,
                                    const float* __restrict__ b2,
                                    const float* __restrict__ W_out,
                                    const float* __restrict__ b_out,
                                    float* __restrict__ out)
{
  __shared__ __align__(16) float zS   [TILE_B][ROWP];
  __shared__ __align__(16) float kcurS[TILE_B][ROWP];
  __shared__ float xdT [3][CDIM][TILE_B];   // [variant][c][b]; xdT[0] doubles as X0 staging
  __shared__ float hbuf[2][TILE_B];

  const int tid  = threadIdx.x;
  const int lane = tid & 31;
  const int wv   = tid >> 5;
  const int b0   = blockIdx.x * TILE_B;
  const int bh   = tid >> 4;                // batch row for h/init/out phases
  const int sh   = tid & 15;

  // ---- per-thread constant weights -> registers ----
  float w1r0[8], w1r1[8];
  #pragma unroll
  for (int m = 0; m < 8; ++m) {
    w1r0[m] = W1[(sh * 8 + m) * 2 + 0];
    w1r1[m] = W1[(sh * 8 + m) * 2 + 1];
  }
  const float b1_0 = b1[0], b1_1 = b1[1];

  const int oo = (tid >> 2) & 3, oq = tid & 3;   // out-projection mapping
  float woutr[32];
  #pragma unroll
  for (int m = 0; m < 32; ++m) woutr[m] = W_out[(oq * 32 + m) * ODIM + oo];
  const float bo = b_out[oo];
  float* const outp = out + ((long)(b0 + bh)) * (LSEQ * ODIM) + oo;

  // ---- W2/b2 A-tiles -> registers (constant across all steps) ----
  // A tile j (16x4 f32, MxK): M = c (16 entries of hidden unit j), K = {W2[0],W2[1],b2,0}
  v2f aW[TPW];
  #pragma unroll
  for (int jj = 0; jj < TPW; ++jj) {
    const int base = (wv * TPW + jj) * CDIM;
    float lo, hi;
    if (lane < 16) { lo = W2[base + lane];        hi = W2[2048 + base + lane]; }
    else           { lo = b2[base + (lane - 16)]; hi = 0.0f; }
    aW[jj].x = lo; aW[jj].y = hi;
  }

  // ---- fused projection: out[:, row, :] = z @ W_out + b_out ----
  auto out_proj = [&](int row) {
    float s = 0.0f;
    const v4f* zp = (const v4f*)&zS[bh][oq * 32];
    #pragma unroll
    for (int g = 0; g < 8; ++g) {
      v4f zv = zp[g];
      #pragma unroll
      for (int e = 0; e < 4; ++e)
        s = __builtin_fmaf(zv[e], woutr[g * 4 + e], s);
    }
    s += __shfl_xor(s, 1);
    s += __shfl_xor(s, 2);
    if (oq == 0) outp[row * ODIM] = s + bo;
  };

  // ---- h = relu((z + alpha*k_cur) @ W1 + b1) ----
  auto h_compute = [&](float alpha, bool use_k) {
    float p0 = 0.0f, p1 = 0.0f;
    const v4f* zp = (const v4f*)&zS[bh][sh * 8];
    const v4f* kp = (const v4f*)&kcurS[bh][sh * 8];
    #pragma unroll
    for (int g = 0; g < 2; ++g) {
      v4f zv = zp[g];
      v4f kv = {};
      if (use_k) kv = kp[g];
      #pragma unroll
      for (int e = 0; e < 4; ++e) {
        float ze = zv[e];
        if (use_k) ze = __builtin_fmaf(alpha, kv[e], ze);
        p0 = __builtin_fmaf(ze, w1r0[g * 4 + e], p0);
        p1 = __builtin_fmaf(ze, w1r1[g * 4 + e], p1);
      }
    }
    #pragma unroll
    for (int w = 8; w >= 1; w >>= 1) { p0 += __shfl_xor(p0, w); p1 += __shfl_xor(p1, w); }
    if (sh == 0) {
      hbuf[0][bh] = fmaxf(p0 + b1_0, 0.0f);
      hbuf[1][bh] = fmaxf(p1 + b1_1, 0.0f);
    }
  };

  // ---- z0 = X0 @ W_init + b_init ----
  xdT[0][bh][sh] = coeffs[((long)(b0 + bh) * TSTEPS) * 64 + sh];   // X0 = a[:,0,:]
  __syncthreads();
  #pragma unroll
  for (int m = 0; m < 8; ++m) {
    const int j = sh * 8 + m;
    float acc = b_init[j];
    #pragma unroll
    for (int c = 0; c < CDIM; ++c)
      acc = __builtin_fmaf(xdT[0][bh][c], W_init[c * HDIM + j], acc);
    zS[bh][j] = acc;
  }
  __syncthreads();
  out_proj(0);

  const float* crow = coeffs + ((long)(b0 + bh) * TSTEPS) * 64 + 16 + sh;
  float xt0[8], xt1[8], xt2[8];

  // ==== RK4 scan over 255 steps ====
  for (int st = 0; st < TSTEPS; ++st) {
    // stage xdot variants, transposed [c][b]; fold in h for vf0 (reads zS only)
    {
      const float vb = crow[0];
      const float vc = crow[16];
      const float vd = crow[32];
      xdT[0][sh][bh] = vb;                                                   // xd0
      xdT[1][sh][bh] = __builtin_fmaf(0.25f, vd, __builtin_fmaf(0.5f, vc, vb)); // xdh
      xdT[2][sh][bh] = vb + vc + vd;                                         // xd1
      if (st + 1 < TSTEPS && sh == 0) {
        __builtin_prefetch(crow + 64, 0, 1);
        __builtin_prefetch(crow + 96, 0, 1);
      }
      crow += 64;
    }
    h_compute(0.0f, false);
    __syncthreads();   // covers xdT + hbuf

    float kacc[TPW];
    #pragma unroll
    for (int i = 0; i < 4; ++i) {
      // ---- g tiles + einsum via f32 WMMA, one hidden unit per 16x16 tile ----
      {
        if (i == 0) {   // hoist all three xdot^T variants into regs once per step
          const int half = lane >> 4, col = lane & 15;
          #pragma unroll
          for (int r = 0; r < 8; ++r) {
            xt0[r] = xdT[0][r + 8 * half][col];
            xt1[r] = xdT[1][r + 8 * half][col];
            xt2[r] = xdT[2][r + 8 * half][col];
          }
        }
        // B matrix = Hpad^T (4x16): K0=h0, K1=h1, K2=1, K3=0
        v2f bm;
        { const float h0v = hbuf[0][lane & 15];
          const float h1v = hbuf[1][lane & 15];
          bm.x = (lane < 16) ? h0v : 1.0f;
          bm.y = (lane < 16) ? h1v : 0.0f; }

        float sreg[TPW];
        #pragma unroll
        for (int jj = 0; jj < TPW; ++jj) {
          v8f acc = {};
          acc = __builtin_amdgcn_wmma_f32_16x16x4_f32(
              false, aW[jj], false, bm, (short)0, acc, false, false);
          float s = 0.0f;
          #pragma unroll
          for (int r = 0; r < 8; ++r) {
            const float x = (i == 0) ? xt0[r] : ((i == 3) ? xt2[r] : xt1[r]);
            s = __builtin_fmaf(fast_tanh(acc[r]), x, s);
          }
          s += __shfl_xor(s, 16);              // combine the two 8-c halves
          sreg[jj] = s;
          kacc[jj] = (i == 0) ? s
                   : __builtin_fmaf((i == 3) ? 1.0f : 2.0f, s, kacc[jj]);
        }
        if (i < 3) {
          if (lane < 16) {                      // owner lane: row of 16 consecutive j
            v4f* kp = (v4f*)&kcurS[lane][wv * TPW];
            #pragma unroll
            for (int g = 0; g < 4; ++g) {
              v4f v = { sreg[g * 4 + 0], sreg[g * 4 + 1],
                        sreg[g * 4 + 2], sreg[g * 4 + 3] };
              kp[g] = v;
            }
          }
        } else {
          if (lane < 16) {                      // fused z += kacc/6 (register accumulator)
            v4f* zp = (v4f*)&zS[lane][wv * TPW];
            #pragma unroll
            for (int g = 0; g < 4; ++g) {
              v4f v = zp[g];
              #pragma unroll
              for (int e = 0; e < 4; ++e)
                v[e] = __builtin_fmaf(kacc[g * 4 + e], (1.0f / 6.0f), v[e]);
              zp[g] = v;
            }
          }
        }
      }
      __syncthreads();
      if (i < 3) {
        h_compute((i < 2) ? 0.5f : 1.0f, true);
        __syncthreads();
      }
    }
    out_proj(st + 1);
  }
}

extern "C" void kernel_launch(void* const* d_in, const int* in_sizes, int n_in,
                              void* d_out, int out_size, void* d_ws, size_t ws_size,
                              hipStream_t stream) {
  (void)in_sizes; (void)n_in; (void)d_ws; (void)ws_size; (void)out_size;
  const float* coeffs = (const float*)d_in[0];
  const float* W_init = (const float*)d_in[1];
  const float* b_init = (const float*)d_in[2];
  const float* W1     = (const float*)d_in[3];
  const float* b1     = (const float*)d_in[4];
  const float* W2     = (const float*)d_in[5];
  const float* b2     = (const float*)d_in[6];
  const float* W_out  = (const float*)d_in[7];
  const float* b_out  = (const float*)d_in[8];
  float* out = (float*)d_out;

  dim3 grid(BATCH / TILE_B);   // 256 blocks, each owns 16 trajectories
  dim3 block(THREADS);         // 8 wave32 waves
  ncde_decoder_kernel<<<grid, block, 0, stream>>>(
      coeffs, W_init, b_init, W1, b1, W2, b2, W_out, b_out, out);
}